// PoleToGaussLegendreGreens_47502338294613
// MI455X (gfx1250) — compile-verified
//
#include <hip/hip_runtime.h>
#include <math.h>

#define LTAU 1000
#define NN   256   // Gauss-Legendre nodes
#define NP   16    // poles
#define NB   16    // batch
#define NWN  256   // Matsubara freqs

typedef float v2f __attribute__((ext_vector_type(2)));
typedef float v8f __attribute__((ext_vector_type(8)));

// ---------------------------------------------------------------------------
// Kernel 1: generate 256-pt Gauss-Legendre nodes/weights in fp64 (Newton on
// the Legendre recurrence), store phi = tan(pi/2 * x) and w as f32 in ws.
// ---------------------------------------------------------------------------
__global__ void gl_setup_kernel(float* __restrict__ phi_out,
                                float* __restrict__ wts_out) {
    const int i = threadIdx.x;           // 256 threads, one node each
    const int N = NN;
    double x = cos(M_PI * (i + 0.75) / (N + 0.5));
    double p0 = 1.0, p1 = x, dp = 1.0;
    for (int it = 0; it < 8; ++it) {
        p0 = 1.0; p1 = x;
        for (int k = 2; k <= N; ++k) {
            double pk = ((2.0 * k - 1.0) * x * p1 - (k - 1.0) * p0) / (double)k;
            p0 = p1; p1 = pk;
        }
        dp = N * (x * p1 - p0) / (x * x - 1.0);
        x -= p1 / dp;
    }
    // derivative at converged x for the weight
    p0 = 1.0; p1 = x;
    for (int k = 2; k <= N; ++k) {
        double pk = ((2.0 * k - 1.0) * x * p1 - (k - 1.0) * p0) / (double)k;
        p0 = p1; p1 = pk;
    }
    dp = N * (x * p1 - p0) / (x * x - 1.0);
    double w = 2.0 / ((1.0 - x * x) * dp * dp);
    phi_out[i] = (float)tan(M_PI * 0.5 * x);
    wts_out[i] = (float)w;
}

// ---------------------------------------------------------------------------
// Kernel 2: main quadrature. grid = (63 tau-tiles, 16 batches), 256 thr = 8
// waves. Wave wv reduces K in [wv*512, wv*512+512) (K = p*256 + n, 4096 wide)
// with V_WMMA_F32_16X16X4_F32: A = 16(tau) x 4(K) reciprocal tile, B = wnum
// broadcast into all 16 columns, C accumulates. Column 0 of D = G_tau tile.
// ---------------------------------------------------------------------------
__global__ void __launch_bounds__(256) gtau_kernel(
        const float* __restrict__ poles_re, const float* __restrict__ poles_im,
        const float* __restrict__ res_re,   const float* __restrict__ res_im,
        const float* __restrict__ phi_g,    const float* __restrict__ wts_g,
        float* __restrict__ out) {
    __shared__ __align__(16) float s_phi[NN];
    __shared__ __align__(16) float s_w[NN];
    __shared__ float s_eps[NP], s_gam[NP], s_ah[NP], s_bh[NP];
    __shared__ float s_part[8 * 16];

    const int b  = blockIdx.y;
    const int l0 = blockIdx.x * 16;
    const int t  = threadIdx.x;

    s_phi[t] = phi_g[t];
    s_w[t]   = wts_g[t];
    if (t < NP) {
        s_eps[t] =  poles_re[b * NP + t];
        s_gam[t] = -poles_im[b * NP + t];
        s_ah[t]  = 0.5f * res_re[b * NP + t];
        s_bh[t]  = 0.5f * res_im[b * NP + t];
    }
    __syncthreads();

    const int lane = t & 31;
    const int wv   = t >> 5;
    const int m    = lane & 15;   // A-matrix row (tau index in tile)
    const int hi   = lane >> 4;   // 0: K0/K1 half, 1: K2/K3 half (ISA layout)

    const float tau  = (float)(l0 + m) * 0.01f;
    const float taum = tau - 10.0f;

    const float2* phi2 = (const float2*)s_phi;
    const float2* w2   = (const float2*)s_w;

    v8f c = {};
    #pragma unroll 1
    for (int pp = 0; pp < 2; ++pp) {
        const int p = wv * 2 + pp;           // pole block for this wave half
        const float eps = s_eps[p], gam = s_gam[p];
        const float ah  = s_ah[p],  bh  = s_bh[p];
        #pragma unroll 4
        for (int ii = 0; ii < 64; ++ii) {
            const int idx = ii * 2 + hi;      // float2 index -> n = 2*idx
            const float2 ph = phi2[idx];
            const float2 wq = w2[idx];
            const float om0 = fmaf(gam, ph.x, eps);
            const float om1 = fmaf(gam, ph.y, eps);
            const float wn0 = wq.x * fmaf(-bh, ph.x, ah);
            const float wn1 = wq.y * fmaf(-bh, ph.y, ah);
            const float a1 = fminf(fmaxf(tau  * om0, -50.f), 50.f);
            const float a2 = fminf(fmaxf(taum * om0, -50.f), 50.f);
            const float a3 = fminf(fmaxf(tau  * om1, -50.f), 50.f);
            const float a4 = fminf(fmaxf(taum * om1, -50.f), 50.f);
            const float d0 = __expf(a1) + __expf(a2);
            const float d1 = __expf(a3) + __expf(a4);
            v2f A, B;
            A.x = __builtin_amdgcn_rcpf(d0);
            A.y = __builtin_amdgcn_rcpf(d1);
            B.x = wn0;
            B.y = wn1;
            // D = A(16x4 f32) x B(4x16 f32) + C, f32 accumulate
            c = __builtin_amdgcn_wmma_f32_16x16x4_f32(
                    false, A, false, B, (short)0, c, false, false);
        }
    }

    // Column 0 of D: lane 0 holds M=0..7, lane 16 holds M=8..15.
    if (lane == 0) {
        #pragma unroll
        for (int j = 0; j < 8; ++j) s_part[wv * 16 + j] = c[j];
    } else if (lane == 16) {
        #pragma unroll
        for (int j = 0; j < 8; ++j) s_part[wv * 16 + 8 + j] = c[j];
    }
    __syncthreads();
    if (t < 16) {
        float s = 0.f;
        #pragma unroll
        for (int w = 0; w < 8; ++w) s += s_part[w * 16 + t];
        const int l = l0 + t;
        if (l < LTAU) out[b * LTAU + l] = s;
    }
}

// ---------------------------------------------------------------------------
// Kernel 3: Matsubara series + asymptotic tails -> G0, overwrite out[b,0].
// One 256-thread block per batch entry, thread t = Matsubara index.
// ---------------------------------------------------------------------------
__global__ void __launch_bounds__(256) g0_kernel(
        const float* __restrict__ poles_re, const float* __restrict__ poles_im,
        const float* __restrict__ res_re,   const float* __restrict__ res_im,
        float* __restrict__ out) {
    __shared__ float s_eps[NP], s_gam[NP], s_a[NP], s_b[NP];
    __shared__ float red[256];
    const int b = blockIdx.x;
    const int t = threadIdx.x;
    if (t < NP) {
        s_eps[t] =  poles_re[b * NP + t];
        s_gam[t] = -poles_im[b * NP + t];
        s_a[t]   =  res_re[b * NP + t];
        s_b[t]   =  res_im[b * NP + t];
    }
    __syncthreads();

    // Tail moment sums over poles: c = -res, then c *= z four times (z=eps-i*gam)
    float S1r = 0.f, S2r = 0.f, S3i = 0.f, S4r = 0.f, S5i = 0.f;
    for (int p = 0; p < NP; ++p) {
        const float zr = s_eps[p], zi = -s_gam[p];
        float cr = -s_a[p], ci = -s_b[p];
        S1r += cr;
        float nr = cr * zr - ci * zi, ni = cr * zi + ci * zr; cr = nr; ci = ni;
        S2r += cr;
        nr = cr * zr - ci * zi; ni = cr * zi + ci * zr; cr = nr; ci = ni;
        S3i += ci;
        nr = cr * zr - ci * zi; ni = cr * zi + ci * zr; cr = nr; ci = ni;
        S4r += cr;
        nr = cr * zr - ci * zi; ni = cr * zi + ci * zr; cr = nr; ci = ni;
        S5i += ci;
    }

    // Re(G_iwn[t]) with tails subtracted; iwn = i*v, v = (2t+1)*pi/beta
    const float v = (2.0f * (float)t + 1.0f) * 3.14159265358979323846f / 10.0f;
    float g = 0.f;
    for (int p = 0; p < NP; ++p) {
        const float x = s_eps[p];
        const float y = -(s_gam[p] + v);        // z - iwn = x + i*y
        const float inv = 1.0f / (x * x + y * y);
        g += (s_a[p] * x + s_b[p] * y) * inv;   // Re(res/(z-iwn))
    }
    const float v2 = v * v, v3 = v2 * v, v4 = v2 * v2, v5 = v4 * v;
    g += S2r / v2 + S3i / v3 - S4r / v4 - S5i / v5;

    red[t] = g;
    __syncthreads();
    for (int s = 128; s > 0; s >>= 1) {
        if (t < s) red[t] += red[t + s];
        __syncthreads();
    }
    if (t == 0) {
        const double Z3 = 1.2020569031595942, Z5 = 1.0369277551433699;
        const double PI = 3.14159265358979323846, B = 10.0;
        const float c0 = -0.5f;
        const float c1 = (float)(-B / 4.0);
        const float c2 = (float)(-7.0 * Z3 * B * B / (4.0 * PI * PI * PI));
        const float c3 = (float)(B * B * B / 48.0);
        const float c4 = (float)(31.0 * Z5 * B * B * B * B /
                                 (16.0 * PI * PI * PI * PI * PI));
        const float G0 = c0 * S1r + c1 * S2r + c2 * S3i + c3 * S4r + c4 * S5i +
                         2.0f * red[0] / 10.0f;
        out[b * LTAU] = G0;
    }
}

extern "C" void kernel_launch(void* const* d_in, const int* in_sizes, int n_in,
                              void* d_out, int out_size, void* d_ws, size_t ws_size,
                              hipStream_t stream) {
    const float* poles_re = (const float*)d_in[0];
    const float* poles_im = (const float*)d_in[1];
    const float* res_re   = (const float*)d_in[2];
    const float* res_im   = (const float*)d_in[3];
    float* out = (float*)d_out;
    float* phi = (float*)d_ws;          // 256 floats
    float* wts = phi + NN;              // 256 floats  (2 KB total in ws)

    gl_setup_kernel<<<1, NN, 0, stream>>>(phi, wts);
    dim3 grid(63, NB);                  // 63 tau-tiles of 16 (covers 1000), 16 batches
    gtau_kernel<<<grid, 256, 0, stream>>>(poles_re, poles_im, res_re, res_im,
                                          phi, wts, out);
    g0_kernel<<<NB, 256, 0, stream>>>(poles_re, poles_im, res_re, res_im, out);
}